// RevIN_17566416241398
// MI455X (gfx1250) — compile-verified
//
#include <hip/hip_runtime.h>

typedef __attribute__((ext_vector_type(2))) float v2f;
typedef __attribute__((ext_vector_type(8))) float v8f;

#define B_      32
#define L_      2048
#define C_      512
#define NCHUNK  8
#define LCHUNK  (L_ / NCHUNK)   /* 256 */
#define EPS_    1e-5f

// ---------------------------------------------------------------------------
// Pass 1: per-wave reduction of 16 channels over an L-chunk using
// V_WMMA_F32_16X16X4_F32 with A = ones.  C row 0 = column sums of B tiles.
//   B tile = 4 consecutive time steps (K) x 16 consecutive channels (N).
//   f32 B layout (mirrors f32 A 16x4): VGPR v, lane-half h -> K = 2*h + v,
//   N = lane & 15.
// ---------------------------------------------------------------------------
__global__ __launch_bounds__(256) void revin_partial(
    const float* __restrict__ x,
    float* __restrict__ part_sum,
    float* __restrict__ part_sq) {
  const int lane  = threadIdx.x & 31;
  const int wave  = threadIdx.x >> 5;
  const int cbase = blockIdx.x * 128 + wave * 16;   // 8 waves * 16 ch = 128 ch/block
  const int b     = blockIdx.y;
  const int chunk = blockIdx.z;
  const int l0    = chunk * LCHUNK;

  const int n     = lane & 15;   // channel within 16-group
  const int khalf = lane >> 4;   // 0 or 1 -> K base 0 or 2

  // A = ones(16x4): every element 1.0 regardless of VGPR striping.
  v2f a; a.x = 1.0f; a.y = 1.0f;

  v8f csum = {};   // accumulates per-channel sum   (replicated across rows)
  v8f csq  = {};   // accumulates per-channel sumsq (replicated across rows)

  const float* base = x + ((size_t)b * L_ + l0) * C_ + (size_t)(cbase + n);

  #pragma unroll 4
  for (int l = 0; l < LCHUNK; l += 4) {
    const float b0 = base[(size_t)(l + 2 * khalf + 0) * C_];  // VGPR0 of B
    const float b1 = base[(size_t)(l + 2 * khalf + 1) * C_];  // VGPR1 of B
    v2f bx;  bx.x  = b0;      bx.y  = b1;
    v2f bx2; bx2.x = b0 * b0; bx2.y = b1 * b1;
    csum = __builtin_amdgcn_wmma_f32_16x16x4_f32(
        false, a, false, bx,  (short)0, csum, false, false);
    csq  = __builtin_amdgcn_wmma_f32_16x16x4_f32(
        false, a, false, bx2, (short)0, csq,  false, false);
  }

  // C/D layout: VGPR0, lanes 0-15 -> (M=0, N=lane).  All rows identical, so
  // lane n holds sum over this L-chunk for channel cbase+n.
  if (lane < 16) {
    const size_t idx = ((size_t)b * C_ + (size_t)(cbase + lane)) * NCHUNK + chunk;
    part_sum[idx] = csum[0];
    part_sq[idx]  = csq[0];
  }
}

// ---------------------------------------------------------------------------
// Pass 2: fold the NCHUNK partials (fixed order -> deterministic), produce
// mean and reciprocal stdev per (b, c).
// ---------------------------------------------------------------------------
__global__ __launch_bounds__(256) void revin_finalize(
    const float* __restrict__ part_sum,
    const float* __restrict__ part_sq,
    float* __restrict__ mean_out,
    float* __restrict__ rstd_out) {
  const int i = blockIdx.x * blockDim.x + threadIdx.x;
  if (i >= B_ * C_) return;
  float s = 0.0f, q = 0.0f;
  #pragma unroll
  for (int c = 0; c < NCHUNK; ++c) {
    s += part_sum[(size_t)i * NCHUNK + c];
    q += part_sq [(size_t)i * NCHUNK + c];
  }
  const float inv_l = 1.0f / (float)L_;
  const float mean  = s * inv_l;
  float var = q * inv_l - mean * mean;   // E[x^2] - E[x]^2
  var = fmaxf(var, 0.0f);
  mean_out[i] = mean;
  rstd_out[i] = 1.0f / sqrtf(var + EPS_);
}

// ---------------------------------------------------------------------------
// Pass 3: elementwise normalize + affine, float4 vectorized along C.
// x re-read here should hit L2 (128 MB data vs 192 MB L2).
// ---------------------------------------------------------------------------
__global__ __launch_bounds__(256) void revin_apply(
    const float* __restrict__ x,
    const float* __restrict__ mean_arr,
    const float* __restrict__ rstd_arr,
    const float* __restrict__ w,
    const float* __restrict__ bias,
    float* __restrict__ out) {
  const size_t idx = (size_t)blockIdx.x * blockDim.x + threadIdx.x;
  // total float4 elements: B*L*(C/4) = 8,388,608
  const size_t c4 = idx & (size_t)(C_ / 4 - 1);     // 0..127
  const size_t bl = idx >> 7;                       // b*L + l
  const size_t b  = bl >> 11;                       // L = 2048
  const int    c  = (int)(c4 * 4);

  const float4 xv = *(const float4*)(x + idx * 4);
  const float4 mv = *(const float4*)(mean_arr + b * C_ + c);
  const float4 rv = *(const float4*)(rstd_arr + b * C_ + c);
  const float4 wv = *(const float4*)(w + c);
  const float4 bv = *(const float4*)(bias + c);

  float4 o;
  o.x = (xv.x - mv.x) * rv.x * wv.x + bv.x;
  o.y = (xv.y - mv.y) * rv.y * wv.y + bv.y;
  o.z = (xv.z - mv.z) * rv.z * wv.z + bv.z;
  o.w = (xv.w - mv.w) * rv.w * wv.w + bv.w;
  *(float4*)(out + idx * 4) = o;
}

// ---------------------------------------------------------------------------
extern "C" void kernel_launch(void* const* d_in, const int* in_sizes, int n_in,
                              void* d_out, int out_size, void* d_ws, size_t ws_size,
                              hipStream_t stream) {
  const float* x    = (const float*)d_in[0];
  const float* aw   = (const float*)d_in[1];
  const float* ab   = (const float*)d_in[2];
  float*       out  = (float*)d_out;

  // Workspace layout (floats):
  //   [0, B*C*NCHUNK)                  part_sum
  //   [B*C*NCHUNK, 2*B*C*NCHUNK)      part_sq
  //   then mean[B*C], rstd[B*C]
  float* ws        = (float*)d_ws;
  float* part_sum  = ws;
  float* part_sq   = ws + (size_t)B_ * C_ * NCHUNK;
  float* mean_arr  = ws + (size_t)2 * B_ * C_ * NCHUNK;
  float* rstd_arr  = mean_arr + (size_t)B_ * C_;

  // Pass 1: (C/128) x B x NCHUNK blocks of 256 threads (8 waves).
  dim3 g1(C_ / 128, B_, NCHUNK);
  revin_partial<<<g1, 256, 0, stream>>>(x, part_sum, part_sq);

  // Pass 2: finalize stats.
  const int stats = B_ * C_;
  revin_finalize<<<(stats + 255) / 256, 256, 0, stream>>>(
      part_sum, part_sq, mean_arr, rstd_arr);

  // Pass 3: normalize + affine, one float4 per thread.
  const size_t total4 = (size_t)B_ * L_ * (C_ / 4);
  revin_apply<<<(unsigned)(total4 / 256), 256, 0, stream>>>(
      x, mean_arr, rstd_arr, aw, ab, out);
}